// GNN_40175124087238
// MI455X (gfx1250) — compile-verified
//
#include <hip/hip_runtime.h>

#define F       128
#define XS_LD   132   // padded LDS stride (floats): bank-conflict-free, 8B-aligned rows
#define WT_LD   132

typedef float v2f __attribute__((ext_vector_type(2)));
typedef float v4f __attribute__((ext_vector_type(4)));
typedef float v8f __attribute__((ext_vector_type(8)));

// ---------------- degree / normalization ----------------

__global__ void gcn_deg_init(float* __restrict__ deg, int n) {
    int i = blockIdx.x * blockDim.x + threadIdx.x;
    if (i < n) deg[i] = 1.0f;               // self-loop weight 1 folded in
}

__global__ void gcn_deg_accum(const long long* __restrict__ col,
                              const float* __restrict__ ew,
                              float* __restrict__ deg, int nE) {
    int e = blockIdx.x * blockDim.x + threadIdx.x;
    if (e < nE) atomicAdd(&deg[col[e]], ew[e]);
}

__global__ void gcn_dinv(const float* __restrict__ deg, float* __restrict__ dinv, int n) {
    int i = blockIdx.x * blockDim.x + threadIdx.x;
    if (i < n) {
        float d = deg[i];
        dinv[i] = (d > 0.0f) ? rsqrtf(d) : 0.0f;
    }
}

// ---------------- WMMA GEMM: h = x @ W, fused self-loop epilogue ----------------
// Block = 256 threads = 8 wave32. Block handles 16 node-rows; wave w handles
// output-feature tile n0 = 16*w. K=128 swept by 32 x v_wmma_f32_16x16x4_f32.

__launch_bounds__(256)
__global__ void gcn_gemm_selfloop(const float* __restrict__ x,
                                  const float* __restrict__ W,
                                  const float* __restrict__ bias,
                                  const float* __restrict__ dinv,
                                  float* __restrict__ h,
                                  float* __restrict__ out, int n) {
    extern __shared__ float lds[];
    float* xs = lds;                       // [16][XS_LD]  x tile (row-major)
    float* wt = lds + 16 * XS_LD;          // [F][WT_LD]   W transposed: wt[nOut][k]

    const int t  = threadIdx.x;
    const int m0 = blockIdx.x * 16;

    // Stage x tile: 16x128 floats, coalesced float4 global loads.
    for (int i = t; i < 16 * F / 4; i += 256) {
        int elem = i * 4;
        int r = elem / F, c = elem % F;
        int gr = m0 + r; if (gr >= n) gr = n - 1;      // clamp tail (n%16==0 normally)
        v4f v = *(const v4f*)&x[(long)gr * F + c];
        *(v4f*)&xs[r * XS_LD + c] = v;
    }
    // Stage W transposed: coalesced float4 loads, scattered 32b LDS stores (one-time).
    for (int i = t; i < F * F / 4; i += 256) {
        int elem = i * 4;
        int r = elem / F, c = elem % F;                // W[r][c..c+3]
        v4f v = *(const v4f*)&W[r * F + c];
        wt[(c + 0) * WT_LD + r] = v.x;
        wt[(c + 1) * WT_LD + r] = v.y;
        wt[(c + 2) * WT_LD + r] = v.z;
        wt[(c + 3) * WT_LD + r] = v.w;
    }
    __syncthreads();

    const int wave = t >> 5;
    const int lane = t & 31;
    const int half = lane >> 4;     // 0: lanes 0-15, 1: lanes 16-31
    const int l16  = lane & 15;
    const int n0   = wave * 16;

    // ISA A-layout (16x4 f32): lane L<16 holds {K0,K1} of row L; L>=16 holds {K2,K3}.
    const float* ap = &xs[l16 * XS_LD + 2 * half];
    // B-layout mirrored: per k-step need W[kbase+2*half + {0,1}][n0+l16] = wt[n][k] pairs.
    const float* bp = &wt[(n0 + l16) * WT_LD + 2 * half];

    v8f acc = {0.f, 0.f, 0.f, 0.f, 0.f, 0.f, 0.f, 0.f};
#pragma unroll
    for (int k = 0; k < F / 4; ++k) {
        v2f a = *(const v2f*)(ap + 4 * k);            // ds_load_b64
        v2f b = *(const v2f*)(bp + 4 * k);            // ds_load_b64
        acc = __builtin_amdgcn_wmma_f32_16x16x4_f32(
            /*neg_a=*/false, a, /*neg_b=*/false, b,
            /*c_mod=*/(short)0, acc, /*reuse_a=*/false, /*reuse_b=*/false);
    }

    // Epilogue: D VGPR g -> M = g + 8*half, N = l16 (ISA 16x16 f32 C/D layout).
#pragma unroll
    for (int g = 0; g < 8; ++g) {
        int m  = m0 + g + 8 * half;
        int nn = n0 + l16;
        if (m < n) {
            float hv = acc[g];
            float di = dinv[m];
            h[(long)m * F + nn]   = hv;                      // L2-resident workspace
            out[(long)m * F + nn] = di * di * hv + bias[nn]; // self-loop + bias
        }
    }
}

// ---------------- edge gather-scale-scatter ----------------
// One wave32 per edge; each lane owns 4 features: float4 gather from h[row],
// 4x global_atomic_add_f32 into out[col]. h/out both fit in 192MB L2.

__launch_bounds__(256)
__global__ void gcn_scatter(const long long* __restrict__ ei,   // [2][E]
                            const float* __restrict__ ew,
                            const float* __restrict__ dinv,
                            const float* __restrict__ h,
                            float* __restrict__ out, int E) {
    int gw = (int)((blockIdx.x * 256u + threadIdx.x) >> 5);     // global wave id = edge
    if (gw >= E) return;
    int lane = threadIdx.x & 31;

    long long r = ei[gw];                 // source node
    long long c = ei[(long long)E + gw];  // target node
    float nrm = dinv[r] * ew[gw] * dinv[c];

    v4f hv = *(const v4f*)&h[r * F + lane * 4];
    float* dst = &out[c * F + lane * 4];
    atomicAdd(dst + 0, nrm * hv.x);
    atomicAdd(dst + 1, nrm * hv.y);
    atomicAdd(dst + 2, nrm * hv.z);
    atomicAdd(dst + 3, nrm * hv.w);
}

// ---------------- launch ----------------

extern "C" void kernel_launch(void* const* d_in, const int* in_sizes, int n_in,
                              void* d_out, int out_size, void* d_ws, size_t ws_size,
                              hipStream_t stream) {
    const float*     x  = (const float*)d_in[0];
    const long long* ei = (const long long*)d_in[1];   // int64 per reference
    const float*     ew = (const float*)d_in[2];
    const float*     W  = (const float*)d_in[3];
    const float*     b  = (const float*)d_in[4];
    float*           out = (float*)d_out;

    const int N = in_sizes[0] / F;      // 50000
    const int E = in_sizes[2];          // 800000

    // Workspace layout: h[N*F] | deg[N] | dinv[N]  (all fully rewritten every call)
    char*  ws   = (char*)d_ws;
    float* h    = (float*)ws;
    float* deg  = (float*)(ws + (size_t)N * F * sizeof(float));
    float* dinv = deg + N;

    gcn_deg_init <<<(N + 255) / 256, 256, 0, stream>>>(deg, N);
    gcn_deg_accum<<<(E + 255) / 256, 256, 0, stream>>>(ei + E, ew, deg, E);
    gcn_dinv     <<<(N + 255) / 256, 256, 0, stream>>>(deg, dinv, N);

    const int mblocks = (N + 15) / 16;  // 3125
    const size_t shbytes = (size_t)(16 * XS_LD + F * WT_LD) * sizeof(float); // ~74 KB < 320 KB/WGP
    gcn_gemm_selfloop<<<mblocks, 256, shbytes, stream>>>(x, W, b, dinv, h, out, N);

    const int eblocks = (E + 7) / 8;    // 8 waves (edges) per block
    gcn_scatter<<<eblocks, 256, 0, stream>>>(ei, ew, dinv, h, out, E);
}